// Attention_41231686042092
// MI455X (gfx1250) — compile-verified
//
#include <hip/hip_runtime.h>
#include <hip/hip_bf16.h>

// ---------------------------------------------------------------------------
// Sparse-attention token-pruning pipeline for MI455X (gfx1250, wave32).
//  * GEMMs: V_WMMA_F32_16X16X4_F32 (exact fp32, matches fp32 reference)
//  * B-tiles staged into LDS by the Tensor Data Mover (TENSOR_LOAD_TO_LDS,
//    inline asm, 2D descriptor w/ LDS padding for bank-conflict-free reads),
//    completion via s_wait_tensorcnt + workgroup barrier.
//  * Exact sparsity threshold: 4-pass radix select on float bit patterns.
// ---------------------------------------------------------------------------

#define B_    16
#define N_    577
#define C_    768
#define H_    12
#define HD    64
#define NP    592          // N padded to multiple of 16 (37 tiles)
#define KKEEP 520          // N - 57 (tokens kept, incl. CLS)
#define NN    (N_ * N_)    // 332929
#define KSEL  166464       // int(N*N*0.5): rank of threshold (descending)
#define SCALE 0.125f       // hd^-0.5

typedef __attribute__((ext_vector_type(2))) float    v2f;
typedef __attribute__((ext_vector_type(8))) float    v8f;
typedef __attribute__((ext_vector_type(4))) unsigned u32x4;
typedef __attribute__((ext_vector_type(8))) unsigned u32x8;

__device__ __forceinline__ v8f wmma4(v2f a, v2f b, v8f c) {
    return __builtin_amdgcn_wmma_f32_16x16x4_f32(false, a, false, b,
                                                 (short)0, c, false, false);
}

// ---------------------------------------------------------------------------
// TDM: async 2D tile load (rows x rowlen fp32) global -> LDS.
// Descriptor per CDNA5 ISA ch.8 (groups 0+1, 2D, data_size=4B).
// pad=true inserts 4 pad dwords after every 256 stored dwords, so the LDS
// linear offset of stored element s is  s + 4*(s>>8).
// Issued by one lane; EXEC is ignored by TDM. Caller does
// s_wait_tensorcnt(0) + __syncthreads() before consuming.
// ---------------------------------------------------------------------------
__device__ __forceinline__ void tdm_load_2d(unsigned lds_byte_off,
                                            const float* gsrc,
                                            unsigned rows, unsigned rowlen,
                                            unsigned row_stride, bool pad) {
    unsigned long long ga = (unsigned long long)(const void*)gsrc;
    u32x4 g0;
    g0[0] = 1u;                                     // count=1 (valid user D#)
    g0[1] = lds_byte_off;                           // lds_addr (bytes)
    g0[2] = (unsigned)ga;                           // global_addr[31:0]
    g0[3] = (unsigned)((ga >> 32) & 0x1ffffffu)     // global_addr[56:32]
          | (2u << 30);                             // type=2 ("image")
    u32x8 g1;
    unsigned w0 = (2u << 16);                       // data_size: 4 bytes
    if (pad) w0 |= (1u << 20)                       // pad_enable
               |  (7u << 22)                        // pad_interval: 256 dwords
               |  (3u << 25);                       // pad_amount: 4 dwords
    g1[0] = w0;                                     // wg_mask=0 (no cluster)
    g1[1] = (rowlen & 0xffffu) << 16;               // tensor_dim0[15:0]
    g1[2] = ((rowlen >> 16) & 0xffffu)              // tensor_dim0[31:16]
          | ((rows & 0xffffu) << 16);               // tensor_dim1[15:0]
    g1[3] = ((rows >> 16) & 0xffffu)                // tensor_dim1[31:16]
          | ((rowlen & 0xffffu) << 16);             // tile_dim0
    g1[4] = rows & 0xffffu;                         // tile_dim1 (tile_dim2=0)
    g1[5] = row_stride;                             // tensor_dim0_stride[31:0]
    g1[6] = 0u;                                     // stride0[47:32], stride1 lo
    g1[7] = 0u;                                     // stride1 hi
    asm volatile("tensor_load_to_lds %0, %1" :: "s"(g0), "s"(g1) : "memory");
}

__device__ __forceinline__ unsigned lds_off32(void* p) {
    // low 32 bits of the generic address == LDS byte offset (ISA aperture map)
    return (unsigned)(unsigned long long)p;
}

// ---------------------------------------------------------------------------
// K0: clear padding (q,k pad rows; vT pad cols) and diag-max buffer.
// ---------------------------------------------------------------------------
__global__ void init_kernel(float* q, float* k, float* v, float* diagmax) {
    const int PAD = NP - N_;                       // 15
    const int tot = B_ * H_ * PAD * HD;
    for (int i = blockIdx.x * blockDim.x + threadIdx.x; i < tot;
         i += gridDim.x * blockDim.x) {
        int d    = i & (HD - 1);
        int t    = i >> 6;
        int prow = t % PAD;
        int bh   = t / PAD;
        q[((size_t)bh * NP + N_ + prow) * HD + d] = 0.f;   // row layout
        k[((size_t)bh * NP + N_ + prow) * HD + d] = 0.f;   // row layout
        v[((size_t)bh * HD + d) * NP + N_ + prow] = 0.f;   // transposed layout
    }
    for (int i = blockIdx.x * blockDim.x + threadIdx.x; i < B_ * (N_ - 1);
         i += gridDim.x * blockDim.x)
        diagmax[i] = 0.f;
}

// ---------------------------------------------------------------------------
// K1: QKV projection. out[m,c] = sum_k x[m,k]*qkv_w[c,k]
// Block = 4 waves sharing one 16-col B tile (TDM-staged, 16x768 fp32 in LDS)
// and covering 4 consecutive 16-row M tiles. q pre-scaled; v stored
// transposed vT[b,h,d,n].
// ---------------------------------------------------------------------------
__global__ void gemm_qkv_kernel(const float* __restrict__ x,
                                const float* __restrict__ w,
                                float* q, float* k, float* v) {
    __shared__ float tb[16 * 780 + 8];             // 768 + 3*4 pad dwords/row
    const int MT = (B_ * N_) / 16;                 // 577
    const int NT = (3 * C_) / 16;                  // 144
    int nt  = blockIdx.x % NT;
    int mtg = blockIdx.x / NT;
    if (threadIdx.x == 0)
        tdm_load_2d(lds_off32(tb), w + (size_t)nt * 16 * C_, 16, C_, C_, true);
    __builtin_amdgcn_s_wait_tensorcnt(0);
    __syncthreads();

    int mt = mtg * 4 + (threadIdx.x >> 5);
    if (mt >= MT) return;
    int lane = threadIdx.x & 31;
    int r    = lane & 15;
    int ko   = (lane < 16) ? 0 : 2;

    const float* arow = x + (size_t)(mt * 16 + r) * C_;
    const float* brow = &tb[r * 780];              // LDS row of this out-col
    v8f acc = {0.f, 0.f, 0.f, 0.f, 0.f, 0.f, 0.f, 0.f};
#pragma unroll 4
    for (int k0 = 0; k0 < C_; k0 += 4) {
        int kk = k0 + ko;
        v2f a = {arow[kk], arow[kk + 1]};
        v2f b = *(const v2f*)&brow[kk + ((kk >> 8) << 2)];
        acc = wmma4(a, b, acc);
    }
    int col   = nt * 16 + r;
    int which = col / C_;
    int cc    = col - which * C_;
    int h     = cc >> 6, d = cc & 63;
#pragma unroll
    for (int i = 0; i < 8; ++i) {
        int mrow = mt * 16 + i + ((lane < 16) ? 0 : 8);
        int b    = mrow / N_;
        int n    = mrow - b * N_;
        if (which == 2)
            v[(((size_t)b * H_ + h) * HD + d) * NP + n] = acc[i];
        else if (which == 1)
            k[(((size_t)b * H_ + h) * NP + n) * HD + d] = acc[i];
        else
            q[(((size_t)b * H_ + h) * NP + n) * HD + d] = acc[i] * SCALE;
    }
}

// ---------------------------------------------------------------------------
// K2: scores per (b,h): S = q @ k^T  ([NPxHD]*[HDxNP] -> [NPxNP], K=64).
// ---------------------------------------------------------------------------
__global__ void gemm_scores_kernel(const float* __restrict__ q,
                                   const float* __restrict__ kmat,
                                   float* __restrict__ attn) {
    const int T = NP / 16;                         // 37
    int bh   = blockIdx.y;
    int tile = blockIdx.x * 4 + (threadIdx.x >> 5);
    if (tile >= T * T) return;
    int lane = threadIdx.x & 31;
    int mt = tile / T, nt = tile - mt * T;
    int r  = lane & 15;
    int ko = (lane < 16) ? 0 : 2;

    const float* arow = q    + ((size_t)bh * NP + mt * 16 + r) * HD;
    const float* brow = kmat + ((size_t)bh * NP + nt * 16 + r) * HD;
    v8f acc = {0.f, 0.f, 0.f, 0.f, 0.f, 0.f, 0.f, 0.f};
#pragma unroll
    for (int k0 = 0; k0 < HD; k0 += 4) {
        v2f a = {arow[k0 + ko], arow[k0 + ko + 1]};
        v2f b = {brow[k0 + ko], brow[k0 + ko + 1]};
        acc = wmma4(a, b, acc);
    }
    float* out = attn + (size_t)bh * NP * NP;
    int ncol = nt * 16 + r;
#pragma unroll
    for (int i = 0; i < 8; ++i) {
        int mrow = mt * 16 + i + ((lane < 16) ? 0 : 8);
        out[(size_t)mrow * NP + ncol] = acc[i];
    }
}

// ---------------------------------------------------------------------------
// K3: row softmax over first N_ cols; zero the pad cols. One wave per row.
// ---------------------------------------------------------------------------
__global__ void softmax_kernel(float* __restrict__ attn) {
    const int lane = threadIdx.x & 31;
    const int wid  = threadIdx.x >> 5;
    long rowg = (long)blockIdx.x * 8 + wid;
    if (rowg >= (long)B_ * H_ * N_) return;
    int bh  = (int)(rowg / N_);
    int row = (int)(rowg - (long)bh * N_);
    float* p = attn + ((size_t)bh * NP + row) * NP;

    float vals[19];
    float mx = -3.0e38f;
#pragma unroll
    for (int t = 0; t < 19; ++t) {
        int j = lane + t * 32;
        float x = (j < N_) ? p[j] : -3.0e38f;
        vals[t] = x;
        mx = fmaxf(mx, x);
    }
#pragma unroll
    for (int o = 16; o > 0; o >>= 1) mx = fmaxf(mx, __shfl_xor(mx, o, 32));
    float s = 0.f;
#pragma unroll
    for (int t = 0; t < 19; ++t) {
        int j = lane + t * 32;
        float e = (j < N_) ? __expf(vals[t] - mx) : 0.f;
        vals[t] = e;
        s += e;
    }
#pragma unroll
    for (int o = 16; o > 0; o >>= 1) s += __shfl_xor(s, o, 32);
    float inv = 1.f / s;
#pragma unroll
    for (int t = 0; t < 19; ++t) {
        int j = lane + t * 32;
        if (j < N_) p[j] = vals[t] * inv;
    }
    if (lane < NP - N_) p[N_ + lane] = 0.f;
}

// ---------------------------------------------------------------------------
// K4: exact radix select of the KSEL-th largest value per (b,h).
// ---------------------------------------------------------------------------
__global__ void select_kernel(const float* __restrict__ attn,
                              float* __restrict__ sigma) {
    int bh = blockIdx.x;
    const float* base = attn + (size_t)bh * NP * NP;
    __shared__ unsigned hist[256];
    __shared__ unsigned s_prefix, s_k;
    if (threadIdx.x == 0) { s_prefix = 0u; s_k = (unsigned)KSEL; }
    __syncthreads();
    for (int pass = 0; pass < 4; ++pass) {
        int shift = 24 - 8 * pass;
        unsigned pm = (pass == 0) ? 0u : (0xFFFFFFFFu << (shift + 8));
        hist[threadIdx.x] = 0u;
        __syncthreads();
        unsigned prefix = s_prefix;
        for (int i = threadIdx.x; i < NN; i += blockDim.x) {
            int r = i / N_, c = i - r * N_;
            unsigned u = __float_as_uint(base[(size_t)r * NP + c]);
            if ((u & pm) == prefix)
                atomicAdd(&hist[(u >> shift) & 255u], 1u);
        }
        __syncthreads();
        if (threadIdx.x == 0) {
            unsigned kk = s_k;
            int byte_ = 0;
            for (int b = 255; b >= 0; --b) {
                unsigned cnt = hist[b];
                if (kk < cnt) { byte_ = b; break; }
                kk -= cnt;
            }
            s_k = kk;
            s_prefix = prefix | ((unsigned)byte_ << shift);
        }
        __syncthreads();
    }
    if (threadIdx.x == 0) sigma[bh] = __uint_as_float(s_prefix);
}

// ---------------------------------------------------------------------------
// K5: threshold in place; fold post-threshold diagonal into max-over-heads.
// ---------------------------------------------------------------------------
__global__ void sparsify_kernel(float* __restrict__ attn,
                                const float* __restrict__ sigma,
                                float* __restrict__ diagmax) {
    int bh = blockIdx.y;
    int i  = blockIdx.x * blockDim.x + threadIdx.x;
    if (i >= NN) return;
    int r = i / N_, c = i - r * N_;
    float sg = sigma[bh];
    float* p = attn + ((size_t)bh * NP + r) * NP + c;
    float val = *p;
    float out = (val >= sg) ? val : 0.f;
    *p = out;
    if (r == c && r > 0) {
        int b = bh / H_;
        atomicMax((int*)&diagmax[b * (N_ - 1) + r - 1], __float_as_int(out));
    }
}

// ---------------------------------------------------------------------------
// K6: stable descending rank -> index_kept (idx[0]=0; idx[rank+1]=tok+1).
// ---------------------------------------------------------------------------
__global__ void rank_kernel(const float* __restrict__ diagmax,
                            int* __restrict__ idx) {
    int b = blockIdx.x;
    __shared__ float keys[N_ - 1];
    for (int i = threadIdx.x; i < N_ - 1; i += blockDim.x)
        keys[i] = diagmax[b * (N_ - 1) + i];
    __syncthreads();
    for (int i = threadIdx.x; i < N_ - 1; i += blockDim.x) {
        float ki = keys[i];
        int rank = 0;
        for (int j = 0; j < N_ - 1; ++j) {
            float kj = keys[j];
            rank += (kj > ki) || ((kj == ki) && (j < i));
        }
        if (rank < KKEEP - 1) idx[b * KKEEP + rank + 1] = i + 1;
    }
    if (threadIdx.x == 0) idx[b * KKEEP] = 0;
}

// ---------------------------------------------------------------------------
// K7: gather x_original rows -> second tuple output.
// ---------------------------------------------------------------------------
__global__ void gather_kernel(const float* __restrict__ x_orig,
                              const int* __restrict__ idx,
                              float* __restrict__ out2) {
    int row = blockIdx.x;                          // b*KKEEP + kk
    int b   = row / KKEEP;
    int src = idx[row];
    const float* s = x_orig + ((size_t)b * N_ + src) * C_;
    float* d = out2 + (size_t)row * C_;
    for (int i = threadIdx.x; i < C_; i += blockDim.x) d[i] = s[i];
}

// ---------------------------------------------------------------------------
// K8: gathered AV GEMM per (b,h).  B = vT tile (16 x NP) TDM-staged in LDS.
//   av[b*KKEEP+kk, h*64+d] = sum_n attn[b,h, idx[kk], n] * vT[b,h,d,n]
// blockIdx.x: [mtg(9) x nt(4)]; 4 waves share nt, cover mt = mtg*4 + wid.
// ---------------------------------------------------------------------------
__global__ void gemm_av_kernel(const float* __restrict__ attn,
                               const float* __restrict__ vT,
                               const int* __restrict__ idx,
                               float* __restrict__ av) {
    __shared__ float tbv[9624];                    // 16 rows x 592 (+pads)
    const int MT = 33;
    int bh  = blockIdx.y;
    int b   = bh / H_;
    int h   = bh - b * H_;
    int nt  = blockIdx.x & 3;
    int mtg = blockIdx.x >> 2;
    if (threadIdx.x == 0)
        tdm_load_2d(lds_off32(tbv), vT + ((size_t)bh * HD + nt * 16) * NP,
                    16, NP, NP, true);
    __builtin_amdgcn_s_wait_tensorcnt(0);
    __syncthreads();

    int mt = mtg * 4 + (threadIdx.x >> 5);
    if (mt >= MT) return;
    int lane = threadIdx.x & 31;
    int r    = lane & 15;
    int ko   = (lane < 16) ? 0 : 2;

    int kkrow = mt * 16 + r;
    int src   = idx[b * KKEEP + ((kkrow < KKEEP) ? kkrow : (KKEEP - 1))];
    const float* arow = attn + ((size_t)bh * NP + src) * NP;
    int d = nt * 16 + r;

    v8f acc = {0.f, 0.f, 0.f, 0.f, 0.f, 0.f, 0.f, 0.f};
#pragma unroll 4
    for (int k0 = 0; k0 < NP; k0 += 4) {
        int kr = k0 + ko;
        v2f a = {arow[kr], arow[kr + 1]};
        int s = r * NP + kr;                       // stored index in tile
        v2f bf = *(const v2f*)&tbv[s + ((s >> 8) << 2)];
        acc = wmma4(a, bf, acc);
    }
#pragma unroll
    for (int i = 0; i < 8; ++i) {
        int kk = mt * 16 + i + ((lane < 16) ? 0 : 8);
        if (kk < KKEEP)
            av[((size_t)b * KKEEP + kk) * C_ + h * HD + d] = acc[i];
    }
}

// ---------------------------------------------------------------------------
// K9: projection GEMM + bias. B tile (16x768) TDM-staged; 4 waves share it.
// ---------------------------------------------------------------------------
__global__ void gemm_proj_kernel(const float* __restrict__ av,
                                 const float* __restrict__ pw,
                                 const float* __restrict__ pb,
                                 float* __restrict__ out) {
    __shared__ float tb[16 * 780 + 8];
    const int NT = C_ / 16;                        // 48
    int nt  = blockIdx.x % NT;
    int mtg = blockIdx.x / NT;
    if (threadIdx.x == 0)
        tdm_load_2d(lds_off32(tb), pw + (size_t)nt * 16 * C_, 16, C_, C_, true);
    __builtin_amdgcn_s_wait_tensorcnt(0);
    __syncthreads();

    int mt   = mtg * 4 + (threadIdx.x >> 5);       // < 520 always (130*4)
    int lane = threadIdx.x & 31;
    int r    = lane & 15;
    int ko   = (lane < 16) ? 0 : 2;

    const float* arow = av + (size_t)(mt * 16 + r) * C_;
    const float* brow = &tb[r * 780];
    v8f acc = {0.f, 0.f, 0.f, 0.f, 0.f, 0.f, 0.f, 0.f};
#pragma unroll 4
    for (int k0 = 0; k0 < C_; k0 += 4) {
        int kk = k0 + ko;
        v2f a = {arow[kk], arow[kk + 1]};
        v2f b = *(const v2f*)&brow[kk + ((kk >> 8) << 2)];
        acc = wmma4(a, b, acc);
    }
    int col  = nt * 16 + r;
    float bias = pb[col];
#pragma unroll
    for (int i = 0; i < 8; ++i) {
        int m = mt * 16 + i + ((lane < 16) ? 0 : 8);
        out[(size_t)m * C_ + col] = acc[i] + bias;
    }
}

// ---------------------------------------------------------------------------
// Launch. Workspace (floats): q,k,vT 3*7,274,496; attn 67,289,088;
// sigma 192; diagmax 9,216; idx 8,320 ints; av 6,389,760  (~383 MB).
// ---------------------------------------------------------------------------
extern "C" void kernel_launch(void* const* d_in, const int* in_sizes, int n_in,
                              void* d_out, int out_size, void* d_ws, size_t ws_size,
                              hipStream_t stream) {
    (void)in_sizes; (void)n_in; (void)out_size; (void)ws_size;
    const float* x      = (const float*)d_in[0];
    const float* x_orig = (const float*)d_in[1];
    const float* qkv_w  = (const float*)d_in[2];
    const float* proj_w = (const float*)d_in[3];
    const float* proj_b = (const float*)d_in[4];
    // d_in[5] = num_prop (always 57; static in this build)

    float* out  = (float*)d_out;
    float* out2 = out + (size_t)B_ * KKEEP * C_;

    const size_t QKV_ELEMS  = (size_t)B_ * H_ * NP * HD;
    const size_t ATTN_ELEMS = (size_t)B_ * H_ * NP * NP;
    float* ws      = (float*)d_ws;
    float* q       = ws;
    float* k       = q + QKV_ELEMS;
    float* v       = k + QKV_ELEMS;                 // transposed: [B,H,HD,NP]
    float* attn    = v + QKV_ELEMS;
    float* sigma   = attn + ATTN_ELEMS;
    float* diagmax = sigma + B_ * H_;
    int*   idx     = (int*)(diagmax + B_ * (N_ - 1));
    float* av      = (float*)(idx + B_ * KKEEP);

    init_kernel<<<256, 256, 0, stream>>>(q, k, v, diagmax);
    gemm_qkv_kernel<<<144 * 145, 128, 0, stream>>>(x, qkv_w, q, k, v);
    gemm_scores_kernel<<<dim3((37 * 37 + 3) / 4, B_ * H_), 128, 0, stream>>>(q, k, attn);
    softmax_kernel<<<(B_ * H_ * N_) / 8, 256, 0, stream>>>(attn);
    select_kernel<<<B_ * H_, 256, 0, stream>>>(attn, sigma);
    sparsify_kernel<<<dim3((NN + 255) / 256, B_ * H_), 256, 0, stream>>>(attn, sigma, diagmax);
    rank_kernel<<<B_, 256, 0, stream>>>(diagmax, idx);
    gather_kernel<<<B_ * KKEEP, 256, 0, stream>>>(x_orig, idx, out2);
    gemm_av_kernel<<<dim3(36, B_ * H_), 128, 0, stream>>>(attn, v, idx, av);
    gemm_proj_kernel<<<48 * 130, 128, 0, stream>>>(av, proj_w, proj_b, out);
}